// Listwise_CE_Loss_25357486915679
// MI455X (gfx1250) — compile-verified
//
#include <hip/hip_runtime.h>
#include <hip/hip_bf16.h>
#include <math.h>

#define PPOS   10
#define NEGN   1000
#define BATCH  4096
#define ROW    1010          // PPOS + NEGN
#define ITEMS  10001
#define GAMMA0 0.1f
#define EPSV   1e-10f

typedef float v2f __attribute__((ext_vector_type(2)));
typedef float v8f __attribute__((ext_vector_type(8)));

// ---------------- copy u -> out_u (out_u is 4B-misaligned: scalar stores) -----
__global__ void copy_u_kernel(const float* __restrict__ u, float* __restrict__ out_u, long n) {
    long i0 = (long)blockIdx.x * blockDim.x + threadIdx.x;
    long stride = (long)gridDim.x * blockDim.x;
    long nv = n >> 2;
    const float4* src = (const float4*)u;
    for (long i = i0; i < nv; i += stride) {
        // prefetch next chunk of the stream (global_prefetch_b8)
        if (i + stride < nv) __builtin_prefetch(&src[i + stride], 0, 1);
        float4 v = src[i];               // aligned B128 load
        long o = i << 2;
        out_u[o + 0] = v.x;              // misaligned dst -> coalesced B32 stores
        out_u[o + 1] = v.y;
        out_u[o + 2] = v.z;
        out_u[o + 3] = v.w;
    }
    if (i0 == 0) {
        for (long i = nv << 2; i < n; ++i) out_u[i] = u[i];
    }
}

// ---------------- per-batch candidate max: max_n neg - min_p pos --------------
__global__ void cand_kernel(const float* __restrict__ pred, float* __restrict__ cand) {
    int b = blockIdx.x;
    int tid = threadIdx.x;
    const float* row = pred + (long)b * ROW;
    float mx = -__FLT_MAX__, mn = __FLT_MAX__;
    for (int i = tid; i < NEGN; i += blockDim.x) mx = fmaxf(mx, row[PPOS + i]);
    if (tid < PPOS) mn = row[tid];
    __shared__ float smx[256], smn[256];
    smx[tid] = mx; smn[tid] = mn;
    __syncthreads();
    for (int s = 128; s > 0; s >>= 1) {
        if (tid < s) {
            smx[tid] = fmaxf(smx[tid], smx[tid + s]);
            smn[tid] = fminf(smn[tid], smn[tid + s]);
        }
        __syncthreads();
    }
    if (tid == 0) cand[b] = smx[0] - smn[0];
}

// ---------------- reduce 4096 candidates -> global max M ---------------------
__global__ void maxred_kernel(const float* __restrict__ cand, float* __restrict__ Mout) {
    __shared__ float s[1024];
    int tid = threadIdx.x;
    float mx = -__FLT_MAX__;
    for (int i = tid; i < BATCH; i += 1024) mx = fmaxf(mx, cand[i]);
    s[tid] = mx;
    __syncthreads();
    for (int k = 512; k > 0; k >>= 1) {
        if (tid < k) s[tid] = fmaxf(s[tid], s[tid + k]);
        __syncthreads();
    }
    if (tid == 0) Mout[0] = s[0];
}

// ---------------- per-batch S,T via v_wmma_f32_16x16x4_f32 -------------------
// One wave per batch row. A-tile carries 64 exp values (or neg*exp); B = ones.
// D[m,n] accumulates row-sums identically in all 16 columns, so
// sum(all 256 accumulator elements) == 16 * sum(all fed values): layout-free.
// Main loop is branch-free (15 x 64 = 960 in-bounds values, B64 loads);
// tail iteration uses clamped indices + value masking (no divergent loads).
__global__ void st_kernel(const float* __restrict__ pred, const float* __restrict__ Mptr,
                          float* __restrict__ S, float* __restrict__ T) {
    int b = blockIdx.x;
    int lane = threadIdx.x;                 // blockDim == 32, one full wave
    float M = Mptr[0];
    const float*  neg  = pred + (long)b * ROW + PPOS;   // 8B-aligned
    const float2* neg2 = (const float2*)neg;
    v8f accE = {}, accT = {};
    v2f ones; ones.x = 1.0f; ones.y = 1.0f;

    // 15 full iterations: n in [0, 960)
    for (int base2 = 0; base2 < 480; base2 += 32) {     // pair index = n/2
        float2 x = neg2[base2 + lane];                  // B64 load
        float e0 = expf(x.x - M);
        float e1 = expf(x.y - M);
        v2f ae; ae.x = e0;       ae.y = e1;
        v2f at; at.x = x.x * e0; at.y = x.y * e1;
        accE = __builtin_amdgcn_wmma_f32_16x16x4_f32(false, ae, false, ones,
                                                     (short)0, accE, false, false);
        accT = __builtin_amdgcn_wmma_f32_16x16x4_f32(false, at, false, ones,
                                                     (short)0, accT, false, false);
    }
    // tail: n in [960, 1024), valid for n < 1000; clamp load index, mask values
    {
        int p2  = 480 + lane;                           // pair index, up to 511
        int p2c = (p2 < 499) ? p2 : 499;                // clamp into bounds
        float2 x = neg2[p2c];
        int n0 = p2 * 2, n1 = n0 + 1;
        float e0 = (n0 < NEGN) ? expf(x.x - M) : 0.0f;
        float e1 = (n1 < NEGN) ? expf(x.y - M) : 0.0f;
        v2f ae; ae.x = e0;       ae.y = e1;
        v2f at; at.x = x.x * e0; at.y = x.y * e1;       // e==0 kills OOB terms
        accE = __builtin_amdgcn_wmma_f32_16x16x4_f32(false, ae, false, ones,
                                                     (short)0, accE, false, false);
        accT = __builtin_amdgcn_wmma_f32_16x16x4_f32(false, at, false, ones,
                                                     (short)0, accT, false, false);
    }

    float pe = 0.0f, pt = 0.0f;
    for (int r = 0; r < 8; ++r) { pe += accE[r]; pt += accT[r]; }
    for (int off = 16; off > 0; off >>= 1) {
        pe += __shfl_xor(pe, off, 32);
        pt += __shfl_xor(pt, off, 32);
    }
    if (lane == 0) {
        S[b] = pe * (1.0f / 16.0f);
        T[b] = pt * (1.0f / 16.0f);
    }
}

// ---------------- scatter: last-write-wins EMA update ------------------------
// Thread i writes iff no j > i shares (uid,pid). Deterministic, no atomics.
__global__ void scatter_kernel(const float* __restrict__ pred, const float* __restrict__ u,
                               const int* __restrict__ user_id, const int* __restrict__ item_id,
                               const float* __restrict__ S, float* __restrict__ out_u) {
    __shared__ int suid[BATCH];             // 16 KB
    for (int j = threadIdx.x; j < BATCH; j += blockDim.x) suid[j] = user_id[j];
    __syncthreads();
    int i = blockIdx.x * blockDim.x + threadIdx.x;   // exactly 40960 threads
    int b = i / PPOS;
    int p = i - b * PPOS;
    int uid = suid[b];
    int pid = item_id[i];
    bool winner = true;
    for (int p2 = p + 1; p2 < PPOS; ++p2)
        if (item_id[b * PPOS + p2] == pid) winner = false;
    for (int b2 = b + 1; b2 < BATCH; ++b2) {
        if (suid[b2] == uid) {
            for (int p2 = 0; p2 < PPOS; ++p2)
                if (item_id[b2 * PPOS + p2] == pid) winner = false;
        }
    }
    if (winner) {
        float pos = pred[(long)b * ROW + p];
        float mean = expf(-pos) * S[b] * (1.0f / (float)NEGN);
        long key = (long)uid * ITEMS + pid;
        out_u[key] = (1.0f - GAMMA0) * u[key] + GAMMA0 * mean;
    }
}

// ---------------- loss partials (gather AFTER scatter) -----------------------
__global__ void loss_kernel(const float* __restrict__ pred, const int* __restrict__ user_id,
                            const int* __restrict__ item_id, const float* __restrict__ S,
                            const float* __restrict__ T, const float* __restrict__ out_u,
                            float* __restrict__ partial) {
    int i = blockIdx.x * blockDim.x + threadIdx.x;   // 40960 threads
    int b = i / PPOS;
    int p = i - b * PPOS;
    float pos = pred[(long)b * ROW + p];
    long key = (long)user_id[b] * ITEMS + item_id[i];
    float denom = out_u[key] + EPSV;
    float c = expf(-pos) * (T[b] - pos * S[b]) / denom;
    __shared__ float s[256];
    int tid = threadIdx.x;
    s[tid] = c;
    __syncthreads();
    for (int k = 128; k > 0; k >>= 1) {
        if (tid < k) s[tid] += s[tid + k];
        __syncthreads();
    }
    if (tid == 0) partial[blockIdx.x] = s[0];
}

__global__ void final_kernel(const float* __restrict__ partial, float* __restrict__ out,
                             int nparts) {
    __shared__ float s[256];
    int tid = threadIdx.x;
    float acc = 0.0f;
    for (int i = tid; i < nparts; i += 256) acc += partial[i];
    s[tid] = acc;
    __syncthreads();
    for (int k = 128; k > 0; k >>= 1) {
        if (tid < k) s[tid] += s[tid + k];
        __syncthreads();
    }
    if (tid == 0) out[0] = s[0] / (float)BATCH;
}

extern "C" void kernel_launch(void* const* d_in, const int* in_sizes, int n_in,
                              void* d_out, int out_size, void* d_ws, size_t ws_size,
                              hipStream_t stream) {
    (void)in_sizes; (void)n_in; (void)out_size; (void)ws_size;
    const float* pred     = (const float*)d_in[0];
    const float* u        = (const float*)d_in[1];
    const int*   user_id  = (const int*)d_in[2];
    const int*   item_id  = (const int*)d_in[3];

    float* out   = (float*)d_out;
    float* out_u = out + 1;                 // u_new region (4B-misaligned)

    float* wsf  = (float*)d_ws;             // ~50 KB used
    float* M    = wsf;
    float* cand = wsf + 8;
    float* S    = wsf + 8 + BATCH;
    float* T    = wsf + 8 + 2 * BATCH;
    float* part = wsf + 8 + 3 * BATCH;

    const long n_u = (long)ITEMS * (long)ITEMS;   // 100,020,001
    const int  nitems  = BATCH * PPOS;            // 40,960
    const int  nblocks = nitems / 256;            // 160

    // Cheap reductions first (don't touch out_u), then the big copy, then
    // scatter -> gather -> loss. Single stream keeps everything ordered.
    cand_kernel   <<<BATCH, 256, 0, stream>>>(pred, cand);
    maxred_kernel <<<1, 1024, 0, stream>>>(cand, M);
    st_kernel     <<<BATCH, 32, 0, stream>>>(pred, M, S, T);
    copy_u_kernel <<<4096, 256, 0, stream>>>(u, out_u, n_u);
    scatter_kernel<<<nblocks, 256, 0, stream>>>(pred, u, user_id, item_id, S, out_u);
    loss_kernel   <<<nblocks, 256, 0, stream>>>(pred, user_id, item_id, S, T, out_u, part);
    final_kernel  <<<1, 256, 0, stream>>>(part, out, nblocks);
}